// ProposalBranch_conf_41626823033500
// MI455X (gfx1250) — compile-verified
//
#include <hip/hip_runtime.h>
#include <math.h>

// ---------------------------------------------------------------------------
// ProposalBranch for MI455X (gfx1250, wave32).
// All GEMM work through V_WMMA_F32_16X16X4_F32 (fp32 in, fp32 accum).
//
// WMMA fp32 lane layouts used (CDNA5 ISA 7.12.2):
//   A (16x4, MxK): lane = 16*half + m ; VGPR0 = K(2*half), VGPR1 = K(2*half+1)
//   B (4x16, KxN): lane = 16*half + n ; VGPR0 = K(2*half), VGPR1 = K(2*half+1)
//   C/D (16x16):   reg r, lane l: M = r + 8*(l>>4), N = l&15
// ---------------------------------------------------------------------------

typedef float f32x2 __attribute__((ext_vector_type(2)));
typedef float f32x8 __attribute__((ext_vector_type(8)));

#define BSZ     4
#define TQ      128        // query/time length of all conv outputs
#define EPS     1e-5f

// ---------------------------------------------------------------------------
// ROI index precompute: Gidx[b*128+n][k] = time index, or -1 if segment empty.
// ---------------------------------------------------------------------------
__global__ void roi_idx_kernel(const float* __restrict__ anchors, // (B,128,2)
                               int K, int Tsrc, float clamp_hi,
                               int* __restrict__ Gidx) {
  int i = blockIdx.x * blockDim.x + threadIdx.x;   // b*128+n
  if (i >= BSZ * TQ) return;
  float a0 = anchors[2 * i + 0];
  float a1 = anchors[2 * i + 1];
  float left  = fminf(fmaxf(a0, 0.0f), clamp_hi);
  float right = fminf(fmaxf(a1, 0.0f), clamp_hi);
  int li = (int)left;                // >= 0 so trunc == floor
  int ri = (int)right;
  int length = min(ri + 1, Tsrc) - li;
  if (length < 1) length = 1;
  bool empty = (right <= left);
  int* g = Gidx + i * K;
  for (int k = 0; k < K; ++k) {
    int idx = li + (k * length) / K;
    idx = min(max(idx, 0), Tsrc - 1);
    g[k] = empty ? -1 : idx;
  }
}

// ---------------------------------------------------------------------------
// Concat copy for feature_lastclass (498 ch @ off 2048) and
// LocResult_feature (2 ch @ off 3570) into the cat buffer (3572 ch).
// ---------------------------------------------------------------------------
__global__ void copy_cat_kernel(const float* __restrict__ lastcls,
                                const float* __restrict__ locres,
                                float* __restrict__ cat) {
  int i = blockIdx.x * blockDim.x + threadIdx.x;
  const int n1 = BSZ * 498 * TQ;
  const int n2 = BSZ * 2 * TQ;
  if (i < n1) {
    int t = i & (TQ - 1);
    int c = (i >> 7) % 498;
    int b = i / (498 * TQ);
    cat[(size_t)b * 3572 * TQ + (size_t)(2048 + c) * TQ + t] = lastcls[i];
  } else if (i < n1 + n2) {
    int j = i - n1;
    int t = j & (TQ - 1);
    int c = (j >> 7) & 1;
    int b = j / (2 * TQ);
    cat[(size_t)b * 3572 * TQ + (size_t)(3570 + c) * TQ + t] = locres[j];
  }
}

// ---------------------------------------------------------------------------
// Boundary max pooling. x (B,C,Tx), seg (B,128,4). Out (B,C,128) at channel
// offset y_choff with row stride 128 (so it can write directly into cat).
// ---------------------------------------------------------------------------
__global__ void bmax_kernel(const float* __restrict__ Xs, int x_bstride, int Tx, int C,
                            const float* __restrict__ seg,
                            float* __restrict__ Y, int y_bstride, int y_choff) {
  int i = blockIdx.x * blockDim.x + threadIdx.x;
  int total = BSZ * C * TQ;
  if (i >= total) return;
  int n = i & (TQ - 1);
  int c = (i >> 7) % C;
  int b = i / (C * TQ);
  const float* s = seg + (size_t)(b * TQ + n) * 4;
  float lo_f, hi_f;
  if (c < (C >> 1)) { lo_f = s[0]; hi_f = s[1]; }
  else              { lo_f = s[2]; hi_f = s[3]; }
  int lo = (int)ceilf(lo_f);  if (lo < 0) lo = 0;
  int hi = (int)floorf(hi_f); if (hi > Tx - 1) hi = Tx - 1;
  float m = 0.0f;
  if (lo <= hi) {
    const float* xr = Xs + (size_t)b * x_bstride + (size_t)c * Tx;
    m = xr[lo];
    for (int t = lo + 1; t <= hi; ++t) m = fmaxf(m, xr[t]);
  }
  Y[(size_t)b * y_bstride + (size_t)(y_choff + c) * TQ + n] = m;
}

// ---------------------------------------------------------------------------
// Fused conv1x1 + GroupNorm + ReLU.
// Block = 8 waves (256 thr) covering a 32-row x 128-col output slab.
// Each wave: one 32x16 slab = 2 M-tiles sharing each B-fragment
// (2 vmem per WMMA). CH=16 -> slab spans 2 groups (2 stat pairs);
// CH=32 -> slab is exactly 1 group (combined stats).
// Grid: (O/32, B).
// ---------------------------------------------------------------------------
template <int CH>
__global__ __launch_bounds__(256)
void convgn_kernel(const float* __restrict__ X, int x_bstride,
                   const float* __restrict__ W, int Cin,
                   const float* __restrict__ bias,
                   const float* __restrict__ gamma,
                   const float* __restrict__ beta,
                   float* __restrict__ Y, int y_bstride, int y_choff) {
  const int tid  = threadIdx.x;
  const int lane = tid & 31;
  const int nt   = tid >> 5;          // 8 n-tiles
  const int half = lane >> 4;
  const int lm   = lane & 15;
  const int cb   = blockIdx.x * 32;   // first output channel of this block
  const int b    = blockIdx.y;

  const float* Xb = X + (size_t)b * x_bstride;
  const float* W0 = W + (size_t)(cb + lm) * Cin;
  const float* W1 = W + (size_t)(cb + 16 + lm) * Cin;
  const int n = nt * 16 + lm;

  f32x8 acc0 = {};
  f32x8 acc1 = {};
#pragma unroll 2
  for (int k0 = 0; k0 < Cin; k0 += 4) {
    int ka = k0 + 2 * half;
    f32x2 a0, a1, bf;
    a0.x = W0[ka];
    a0.y = W0[ka + 1];
    a1.x = W1[ka];
    a1.y = W1[ka + 1];
    bf.x = Xb[(size_t)ka * TQ + n];
    bf.y = Xb[(size_t)(ka + 1) * TQ + n];
    acc0 = __builtin_amdgcn_wmma_f32_16x16x4_f32(false, a0, false, bf,
                                                 (short)0, acc0, false, false);
    acc1 = __builtin_amdgcn_wmma_f32_16x16x4_f32(false, a1, false, bf,
                                                 (short)0, acc1, false, false);
  }

  // bias + per-M-tile moments
  float v0[8], v1[8];
  float ls0 = 0.0f, lq0 = 0.0f, ls1 = 0.0f, lq1 = 0.0f;
#pragma unroll
  for (int r = 0; r < 8; ++r) {
    int rr = r + 8 * half;
    float a = acc0[r] + bias[cb + rr];
    float c = acc1[r] + bias[cb + 16 + rr];
    v0[r] = a; ls0 += a; lq0 += a * a;
    v1[r] = c; ls1 += c; lq1 += c * c;
  }

  __shared__ float s0[256], q0[256], s1[256], q1[256];
  __shared__ float sm0, sr0, sm1, sr1;
  s0[tid] = ls0; q0[tid] = lq0; s1[tid] = ls1; q1[tid] = lq1;
  __syncthreads();
  for (int off = 128; off > 0; off >>= 1) {
    if (tid < off) {
      s0[tid] += s0[tid + off]; q0[tid] += q0[tid + off];
      s1[tid] += s1[tid + off]; q1[tid] += q1[tid + off];
    }
    __syncthreads();
  }
  if (tid == 0) {
    if (CH == 32) {                       // one group: combined stats
      const float invN = 1.0f / (32.0f * TQ);
      float m  = (s0[0] + s1[0]) * invN;
      float va = (q0[0] + q1[0]) * invN - m * m;
      float rs = rsqrtf(va + EPS);
      sm0 = m; sm1 = m; sr0 = rs; sr1 = rs;
    } else {                              // two groups: separate stats
      const float invN = 1.0f / (16.0f * TQ);
      float m0  = s0[0] * invN;
      float va0 = q0[0] * invN - m0 * m0;
      float m1  = s1[0] * invN;
      float va1 = q1[0] * invN - m1 * m1;
      sm0 = m0; sr0 = rsqrtf(va0 + EPS);
      sm1 = m1; sr1 = rsqrtf(va1 + EPS);
    }
  }
  __syncthreads();
  const float m0 = sm0, r0 = sr0, m1 = sm1, r1 = sr1;

#pragma unroll
  for (int r = 0; r < 8; ++r) {
    int rr = r + 8 * half;
    int ch0 = cb + rr;
    int ch1 = cb + 16 + rr;
    float y0 = fmaxf((v0[r] - m0) * r0 * gamma[ch0] + beta[ch0], 0.0f);
    float y1 = fmaxf((v1[r] - m1) * r1 * gamma[ch1] + beta[ch1], 0.0f);
    Y[(size_t)b * y_bstride + (size_t)(y_choff + ch0) * TQ + n] = y0;
    Y[(size_t)b * y_bstride + (size_t)(y_choff + ch1) * TQ + n] = y1;
  }
}

// ---------------------------------------------------------------------------
// Materialize the gathered ROI B-matrix in WMMA-pair-interleaved layout:
//   Bm[((b*(CK/2) + kl/2)*128 + n)*2 + (kl&1)] = data[b, n, c(kl), k(kl)]
// (kl = c*K + k). Emptiness folded in as exact zeros. One thread per element,
// writes fully coalesced; the 16x redundant per-p-slab gathers of the GEMM
// become a single pass here.
// ---------------------------------------------------------------------------
template <int K, int KSHIFT>
__global__ void bmat_kernel(const float* __restrict__ Src, int src_bstride, int Tsrc,
                            const int* __restrict__ Gidx, int CK,
                            float* __restrict__ Bm) {
  int i = blockIdx.x * blockDim.x + threadIdx.x;
  int total = BSZ * CK * TQ;
  if (i >= total) return;
  int j    = i & 1;
  int n    = (i >> 1) & (TQ - 1);
  int rest = i >> 8;
  int kp   = rest % (CK / 2);
  int b    = rest / (CK / 2);
  int kl   = 2 * kp + j;
  int c    = kl >> KSHIFT;
  int k    = kl & (K - 1);
  int t    = Gidx[(size_t)(b * TQ + n) * K + k];
  float v  = (t >= 0) ? Src[(size_t)b * src_bstride + (size_t)c * Tsrc + t] : 0.0f;
  Bm[i] = v;
}

// ---------------------------------------------------------------------------
// Dense GEMM on the materialized B: out[b,p,n] = sum_kl W[p,kl]*Bm[b,kl,n] + bias[p]
// Each wave: (16*MT) x 16 slab; B-fragment = one aligned global_load_b64.
// Grid: B * (512/(16*MT)) * 8 waves, 8 waves/block.
// ---------------------------------------------------------------------------
template <int MT>
__global__ __launch_bounds__(256)
void roigemm_kernel(const float* __restrict__ Bm,
                    const float* __restrict__ Wt, int CK,
                    const float* __restrict__ bias,
                    float* __restrict__ Y, int y_bstride, int y_choff) {
  const int tid  = threadIdx.x;
  const int lane = tid & 31;
  const int wave = tid >> 5;
  const int half = lane >> 4;
  const int lm   = lane & 15;

  const int TILES_PER_B = (512 / (16 * MT)) * 8;
  const int gw  = blockIdx.x * 8 + wave;
  const int b   = gw / TILES_PER_B;
  const int rem = gw % TILES_PER_B;
  const int ps  = rem >> 3;                // p-slab (16*MT rows)
  const int nt  = rem & 7;                 // 8 n-tiles

  const int n = nt * 16 + lm;
  const float* Bb = Bm + (size_t)b * (CK / 2) * (2 * TQ) + (size_t)n * 2;

  const float* Wp[MT];
#pragma unroll
  for (int m = 0; m < MT; ++m)
    Wp[m] = Wt + (size_t)(ps * (16 * MT) + m * 16 + lm) * CK;

  f32x8 acc[MT];
#pragma unroll
  for (int m = 0; m < MT; ++m) acc[m] = (f32x8){};

#pragma unroll 2
  for (int k0 = 0; k0 < CK; k0 += 4) {
    const int kp = (k0 >> 1) + half;
    f32x2 bf = *(const f32x2*)(Bb + (size_t)kp * (2 * TQ));
#pragma unroll
    for (int m = 0; m < MT; ++m) {
      f32x2 af;
      af.x = Wp[m][k0 + 2 * half];
      af.y = Wp[m][k0 + 2 * half + 1];
      acc[m] = __builtin_amdgcn_wmma_f32_16x16x4_f32(false, af, false, bf,
                                                     (short)0, acc[m], false, false);
    }
  }

#pragma unroll
  for (int m = 0; m < MT; ++m) {
#pragma unroll
    for (int r = 0; r < 8; ++r) {
      int p = ps * (16 * MT) + m * 16 + r + 8 * half;
      Y[(size_t)b * y_bstride + (size_t)(y_choff + p) * TQ + n] = acc[m][r] + bias[p];
    }
  }
}

// ---------------------------------------------------------------------------
// Orchestration.
// ---------------------------------------------------------------------------
extern "C" void kernel_launch(void* const* d_in, const int* in_sizes, int n_in,
                              void* d_out, int out_size, void* d_ws, size_t ws_size,
                              hipStream_t stream) {
  const float* feature   = (const float*)d_in[0];   // (4,512,128)
  const float* flf       = (const float*)d_in[1];   // (4,512,256)
  const float* segments  = (const float*)d_in[2];   // (4,128,4)
  const float* fsegments = (const float*)d_in[3];   // (4,128,4)
  const float* lastcls   = (const float*)d_in[4];   // (4,498,128)
  const float* anchor    = (const float*)d_in[5];   // (4,128,2)
  const float* fanchor   = (const float*)d_in[6];   // (4,128,2)
  /* d_in[7] = order (always 0 -> K1 = KER[1] = 8) */
  const float* locres    = (const float*)d_in[8];   // (4,2,128)
  const float* w_cur  = (const float*)d_in[9];
  const float* b_cur  = (const float*)d_in[10];
  const float* g_cur  = (const float*)d_in[11];
  const float* bt_cur = (const float*)d_in[12];
  const float* w_lr   = (const float*)d_in[13];
  const float* b_lr   = (const float*)d_in[14];
  const float* g_lr   = (const float*)d_in[15];
  const float* bt_lr  = (const float*)d_in[16];
  const float* w_roi  = (const float*)d_in[17];
  const float* b_roi  = (const float*)d_in[18];
  const float* g_roi  = (const float*)d_in[19];
  const float* bt_roi = (const float*)d_in[20];
  const float* w_bcp0 = (const float*)d_in[21];     // (512,512,16)
  const float* b_bcp0 = (const float*)d_in[22];
  const float* w_bcpl = (const float*)d_in[23];     // (512,1024,8)
  const float* b_bcpl = (const float*)d_in[24];
  const float* w_prop = (const float*)d_in[25];     // (512,3572)
  const float* b_prop = (const float*)d_in[26];
  const float* g_prop = (const float*)d_in[27];
  const float* bt_prop = (const float*)d_in[28];

  float* out  = (float*)d_out;                  // (4,512,128)
  float* feat = out + (size_t)BSZ * 512 * TQ;   // (4,1024,128)

  const int    CK     = 8192;                         // both ROI GEMMs
  const size_t CAT_SZ = (size_t)BSZ * 3572 * TQ;      // 1,828,864 floats
  float* cat      = (float*)d_ws;
  float* pooltmp  = cat + CAT_SZ;                     // (4,512,128)
  float* framecon = pooltmp + (size_t)BSZ * 512 * TQ; // (4,512,128)
  int*   gidx1    = (int*)(framecon + (size_t)BSZ * 512 * TQ);  // 4*128*8
  int*   gidx2    = gidx1 + BSZ * TQ * 8;                       // 4*128*16
  float* Bmat     = (float*)(gidx2 + BSZ * TQ * 16);  // (4, CK/2, 128, 2) reused

  // 1) ROI index tables
  roi_idx_kernel<<<2, 256, 0, stream>>>(anchor, 8, 128, 128.0f, gidx1);
  roi_idx_kernel<<<2, 256, 0, stream>>>(fanchor, 16, 256, 256.0f, gidx2);

  // 2) pass-through concat channels
  copy_cat_kernel<<<(BSZ * 500 * TQ + 255) / 256, 256, 0, stream>>>(lastcls, locres, cat);

  // 3) fm_short = cgr(feature, w_cur)  -> cat channels [1536,2048)
  convgn_kernel<16><<<dim3(16, BSZ), 256, 0, stream>>>(
      feature, 512 * TQ, w_cur, 512, b_cur, g_cur, bt_cur, cat, 3572 * TQ, 1536);

  // 4) feat = cgr(feature, w_lr) -> second output, (4,1024,128)
  convgn_kernel<32><<<dim3(32, BSZ), 256, 0, stream>>>(
      feature, 512 * TQ, w_lr, 512, b_lr, g_lr, bt_lr, feat, 1024 * TQ, 0);

  // 5) prop_feature = bmax(feat, segments) -> cat [512,1536)
  bmax_kernel<<<(BSZ * 1024 * TQ + 255) / 256, 256, 0, stream>>>(
      feat, 1024 * TQ, 128, 1024, segments, cat, 3572 * TQ, 512);

  // 6) raw prop_roi = bmax(frame_level_feature, frame_segments) -> pooltmp
  bmax_kernel<<<(BSZ * 512 * TQ + 255) / 256, 256, 0, stream>>>(
      flf, 512 * 256, 256, 512, fsegments, pooltmp, 512 * TQ, 0);

  // 7) prop_roi = cgr(pooltmp, w_roi) -> cat [0,512)
  convgn_kernel<16><<<dim3(16, BSZ), 256, 0, stream>>>(
      pooltmp, 512 * TQ, w_roi, 512, b_roi, g_roi, bt_roi, cat, 3572 * TQ, 0);

  // 8) con_feature = roiGEMM(feat, gidx1, w_bcpl) + b -> cat [2546,3058)
  bmat_kernel<8, 3><<<(BSZ * CK * TQ + 255) / 256, 256, 0, stream>>>(
      feat, 1024 * TQ, 128, gidx1, CK, Bmat);
  roigemm_kernel<2><<<64, 256, 0, stream>>>(
      Bmat, w_bcpl, CK, b_bcpl, cat, 3572 * TQ, 2546);

  // 9) frame_con = roiGEMM(flf, gidx2, w_bcp0) + b -> framecon (Bmat reused)
  bmat_kernel<16, 4><<<(BSZ * CK * TQ + 255) / 256, 256, 0, stream>>>(
      flf, 512 * 256, 256, gidx2, CK, Bmat);
  roigemm_kernel<2><<<64, 256, 0, stream>>>(
      Bmat, w_bcp0, CK, b_bcp0, framecon, 512 * TQ, 0);

  // 10) prop_con = cgr(framecon, w_roi) -> cat [3058,3570)
  convgn_kernel<16><<<dim3(16, BSZ), 256, 0, stream>>>(
      framecon, 512 * TQ, w_roi, 512, b_roi, g_roi, bt_roi, cat, 3572 * TQ, 3058);

  // 11) out = cgr(cat, w_prop), C_in = 3572 -> first output
  convgn_kernel<16><<<dim3(16, BSZ), 256, 0, stream>>>(
      cat, 3572 * TQ, w_prop, 3572, b_prop, g_prop, bt_prop, out, 512 * TQ, 0);
}